// Gaussian_Classifier_11519102287933
// MI455X (gfx1250) — compile-verified
//
#include <hip/hip_runtime.h>
#include <math.h>
#include <stdint.h>

typedef float v2f __attribute__((ext_vector_type(2)));
typedef float v8f __attribute__((ext_vector_type(8)));

#define B_    8
#define N_    512
#define NODE_ 3
#define H_    64
#define K_    10
#define KSPLIT 4            // waves per workgroup in wtot kernel (k-range split)

#if __has_builtin(__builtin_amdgcn_global_load_async_to_lds_b128) && \
    __has_builtin(__builtin_amdgcn_s_wait_asynccnt)
#define USE_ASYNC_LDS 1
// exact parameter types per hipcc diagnostic: (v4i32 AS1*, v4i32 AS3*, imm, imm)
typedef int v4i_vs __attribute__((vector_size(16)));
typedef __attribute__((address_space(1))) v4i_vs* async_gptr_t;
typedef __attribute__((address_space(3))) v4i_vs* async_lptr_t;
#else
#define USE_ASYNC_LDS 0
#endif

// ---------------- kernel 1: h0 = pad(x, H) ----------------
__global__ void init_h_kernel(const float* __restrict__ x, float* __restrict__ h) {
    int tid = blockIdx.x * blockDim.x + threadIdx.x;
    if (tid >= B_ * N_ * H_) return;
    int c = tid & (H_ - 1);
    int i = (tid >> 6) & (N_ - 1);
    int b = tid >> 15;
    h[tid] = (c < NODE_) ? x[(b * N_ + i) * NODE_ + c] : 0.0f;
}

// ---------------- kernel 2: y = h @ fn_w^T + fn_b ----------------
// One wave32 per (batch, 16-row tile). 16x64 tile in 4 x v8f accumulators.
__global__ void y_gemm_kernel(const float* __restrict__ h,
                              const float* __restrict__ fn_w,
                              const float* __restrict__ fn_b,
                              float* __restrict__ y) {
    int lane = threadIdx.x;            // 0..31
    int it   = blockIdx.x;             // row tile
    int b    = blockIdx.y;             // batch
    int i0   = it * 16;
    const float* hb = h + (size_t)b * N_ * H_;
    float*       yb = y + (size_t)b * N_ * H_;
    int m  = lane & 15;
    int hi = lane >> 4;                // 0 or 1 (selects K pair)

    // pull the (shared, reused-by-every-block) weight matrix toward the caches
    __builtin_prefetch(&fn_w[lane * 128], 0, 3);

    v8f acc[4] = {};
    for (int kk = 0; kk < H_ / 4; ++kk) {
        int c0 = kk * 4 + (hi << 1);   // A/B K index for VGPR0 of this lane
        v2f a;
        a.x = hb[(i0 + m) * H_ + c0];
        a.y = hb[(i0 + m) * H_ + c0 + 1];
        #pragma unroll
        for (int nt = 0; nt < 4; ++nt) {
            int n = nt * 16 + m;
            v2f bb;                    // B(k,n) = fn_w[n][k]
            bb.x = fn_w[n * H_ + c0];
            bb.y = fn_w[n * H_ + c0 + 1];
            acc[nt] = __builtin_amdgcn_wmma_f32_16x16x4_f32(
                false, a, false, bb, (short)0, acc[nt], false, false);
        }
    }
    #pragma unroll
    for (int nt = 0; nt < 4; ++nt) {
        float bias = fn_b[nt * 16 + m];
        #pragma unroll
        for (int r = 0; r < 8; ++r)
            yb[(i0 + r + 8 * hi) * H_ + nt * 16 + m] = acc[nt][r] + bias;
    }
}

// ---------------- kernel 3: h_next = (exp-kernel Wtot) @ y, fused ----------------
// 4 waves per (batch, 16-row tile); each wave owns a 128-deep k-range and a
// private LDS slab (no barriers in the hot loop), partials reduced through LDS.
// Wtot is generated in registers directly in the WMMA A-operand layout.
__global__ void wtot_gemm_kernel(const float* __restrict__ hcur,
                                 const float* __restrict__ y,
                                 const float* __restrict__ mu,
                                 const float* __restrict__ sigma,
                                 const float* __restrict__ kw,
                                 float* __restrict__ hnext) {
    __shared__ float lds_y[KSPLIT][16 * H_];   // per-wave 16x64 y slab (4 KB each)
    __shared__ float lds_pos[KSPLIT][32];      // per-wave 16 k-positions
    int tid  = threadIdx.x;
    int lane = tid & 31;
    int wv   = tid >> 5;                       // wave id 0..3
    int it = blockIdx.x, b = blockIdx.y;
    int i0 = it * 16;
    const float* hb  = hcur  + (size_t)b * N_ * H_;
    const float* yb  = y     + (size_t)b * N_ * H_;
    float*       hnb = hnext + (size_t)b * N_ * H_;
    int m  = lane & 15;
    int hi = lane >> 4;

    // Gaussian mixture params, per-lane registers
    float s0[K_], s1[K_], sm0[K_], sm1[K_], cst[K_], w[K_];
    #pragma unroll
    for (int j = 0; j < K_; ++j) {
        float sj0 = sigma[j * 2 + 0], sj1 = sigma[j * 2 + 1];
        float mj0 = mu[j * 2 + 0],    mj1 = mu[j * 2 + 1];
        s0[j] = sj0; s1[j] = sj1;
        sm0[j] = sj0 * mj0; sm1[j] = sj1 * mj1;
        cst[j] = sj0 * mj0 * mj0 + sj1 * mj1 * mj1;
        w[j] = kw[j];
    }
    float pix = hb[(i0 + m) * H_ + 0];
    float piy = hb[(i0 + m) * H_ + 1];

    v8f acc[4] = {};
    const int kbeg = wv * (N_ / KSPLIT);
    const int kend = kbeg + (N_ / KSPLIT);
    for (int k0 = kbeg; k0 < kend; k0 += 16) {
        if (lane < 16) {                       // stage 16 k-positions (wave-private)
            lds_pos[wv][lane * 2 + 0] = hb[(k0 + lane) * H_ + 0];
            lds_pos[wv][lane * 2 + 1] = hb[(k0 + lane) * H_ + 1];
        }
#if USE_ASYNC_LDS
        #pragma unroll
        for (int p = 0; p < 8; ++p) {          // async-stage 16x64 y slab, b128/lane
            int idx = p * 32 + lane;           // 0..255
            int row = idx >> 4;
            int c4  = (idx & 15) << 2;
            const float* gp = &yb[(k0 + row) * H_ + c4];
            float*       lp = &lds_y[wv][row * H_ + c4];
            // generic->AS casts via integer: LDS generic addr[31:0] == LDS offset
            __builtin_amdgcn_global_load_async_to_lds_b128(
                (async_gptr_t)(unsigned long long)(uintptr_t)gp,
                (async_lptr_t)(unsigned int)(uintptr_t)lp,
                0, 0);
        }
        __builtin_amdgcn_s_wait_asynccnt(0);
#else
        #pragma unroll
        for (int p = 0; p < 8; ++p) {          // stage 16x64 y slab, float4 coalesced
            int idx = p * 32 + lane;
            int row = idx >> 4;
            int c4  = (idx & 15) << 2;
            float4 v = *(const float4*)&yb[(k0 + row) * H_ + c4];
            *(float4*)&lds_y[wv][row * H_ + c4] = v;
        }
#endif
        // no cross-wave barrier needed: slab is wave-private, DS ops in-order per wave

        #pragma unroll
        for (int kk = 0; kk < 4; ++kk) {
            int c0 = kk * 4 + (hi << 1);       // K index of this lane's A VGPR0
            float wvv[2];
            #pragma unroll
            for (int t = 0; t < 2; ++t) {      // W(i0+m, k0+c0+t) = sum_j kw_j*exp(...)
                float ux = lds_pos[wv][(c0 + t) * 2 + 0] - pix;
                float uy = lds_pos[wv][(c0 + t) * 2 + 1] - piy;
                float ax = 0.0f;
                #pragma unroll
                for (int j = 0; j < K_; ++j) {
                    float e = s0[j] * ux * ux + s1[j] * uy * uy
                            - 2.0f * (sm0[j] * ux + sm1[j] * uy) + cst[j];
                    ax += w[j] * __expf(e);
                }
                wvv[t] = ax;
            }
            v2f a; a.x = wvv[0]; a.y = wvv[1];
            #pragma unroll
            for (int nt = 0; nt < 4; ++nt) {
                int n = nt * 16 + m;
                v2f bb;
                bb.x = lds_y[wv][c0 * H_ + n];
                bb.y = lds_y[wv][(c0 + 1) * H_ + n];
                acc[nt] = __builtin_amdgcn_wmma_f32_16x16x4_f32(
                    false, a, false, bb, (short)0, acc[nt], false, false);
            }
        }
    }

    // --- reduce the KSPLIT partial 16x64 tiles through LDS ---
    if (wv != 0) {
        #pragma unroll
        for (int nt = 0; nt < 4; ++nt)
            #pragma unroll
            for (int r = 0; r < 8; ++r)
                lds_y[wv][(r + 8 * hi) * H_ + nt * 16 + m] = acc[nt][r];
    }
    __syncthreads();
    if (wv == 0) {
        #pragma unroll
        for (int nt = 0; nt < 4; ++nt)
            #pragma unroll
            for (int r = 0; r < 8; ++r) {
                int off = (r + 8 * hi) * H_ + nt * 16 + m;
                float s = acc[nt][r] + lds_y[1][off] + lds_y[2][off] + lds_y[3][off];
                hnb[(i0 + r + 8 * hi) * H_ + nt * 16 + m] = s;
            }
    }
}

// ---------------- kernel 4: head (fc1+relu, mean over N, fc2, log_softmax) ----------------
__global__ void head_kernel(const float* __restrict__ h,
                            const float* __restrict__ fc1_w,
                            const float* __restrict__ fc1_b,
                            const float* __restrict__ fc2_w,
                            const float* __restrict__ fc2_b,
                            float* __restrict__ out) {
    __shared__ float msum[50];
    int b = blockIdx.x;
    int tid = threadIdx.x;             // 0..255
    const float* hb = h + (size_t)b * N_ * H_;
    if (tid < 50) msum[tid] = 0.0f;
    __syncthreads();

    float acc[50];
    #pragma unroll
    for (int c = 0; c < 50; ++c) acc[c] = 0.0f;
    for (int i = tid; i < N_; i += 256) {
        const float* hr = hb + i * H_;
        for (int c = 0; c < 50; ++c) {
            float d = fc1_b[c];
            const float* wr = fc1_w + c * H_;
            #pragma unroll
            for (int k = 0; k < H_; ++k) d += hr[k] * wr[k];
            acc[c] += fmaxf(d, 0.0f);
        }
    }
    for (int c = 0; c < 50; ++c) atomicAdd(&msum[c], acc[c]);  // ds_add_f32
    __syncthreads();

    if (tid == 0) {
        float z[10];
        float mx = -1e30f;
        for (int o = 0; o < 10; ++o) {
            float s = fc2_b[o];
            for (int c = 0; c < 50; ++c)
                s += (msum[c] * (1.0f / N_)) * fc2_w[o * 50 + c];
            z[o] = s;
            mx = fmaxf(mx, s);
        }
        float se = 0.0f;
        for (int o = 0; o < 10; ++o) se += __expf(z[o] - mx);
        float lse = mx + __logf(se);
        for (int o = 0; o < 10; ++o) out[b * 10 + o] = z[o] - lse;
    }
}

extern "C" void kernel_launch(void* const* d_in, const int* in_sizes, int n_in,
                              void* d_out, int out_size, void* d_ws, size_t ws_size,
                              hipStream_t stream) {
    const float* x     = (const float*)d_in[0];
    const float* fn_w  = (const float*)d_in[1];
    const float* fn_b  = (const float*)d_in[2];
    const float* mu    = (const float*)d_in[3];
    const float* sigma = (const float*)d_in[4];
    const float* kw    = (const float*)d_in[5];
    const float* fc1_w = (const float*)d_in[6];
    const float* fc1_b = (const float*)d_in[7];
    const float* fc2_w = (const float*)d_in[8];
    const float* fc2_b = (const float*)d_in[9];
    float* out = (float*)d_out;

    // workspace: h ping, h pong, y  (1 MB each)
    float* h0   = (float*)d_ws;
    float* h1   = h0 + (size_t)B_ * N_ * H_;
    float* ybuf = h1 + (size_t)B_ * N_ * H_;

    init_h_kernel<<<(B_ * N_ * H_ + 255) / 256, 256, 0, stream>>>(x, h0);

    float* cur = h0;
    float* nxt = h1;
    for (int iter = 0; iter < 2; ++iter) {
        y_gemm_kernel<<<dim3(N_ / 16, B_), 32, 0, stream>>>(cur, fn_w, fn_b, ybuf);
        wtot_gemm_kernel<<<dim3(N_ / 16, B_), 32 * KSPLIT, 0, stream>>>(cur, ybuf, mu, sigma, kw, nxt);
        float* t = cur; cur = nxt; nxt = t;
    }
    head_kernel<<<B_, 256, 0, stream>>>(cur, fc1_w, fc1_b, fc2_w, fc2_b, out);
}